// Invertible_1x1_Conv_54735063220402
// MI455X (gfx1250) — compile-verified
//
#include <hip/hip_runtime.h>
#include <hip/hip_bf16.h>
#include <math.h>

typedef __attribute__((ext_vector_type(2))) float v2f;
typedef __attribute__((ext_vector_type(8))) float v8f;

#define B_   16
#define C_   192
#define T_   4096
#define G_   48      // C / NUM_SPLIT
#define CHUNK 512    // t-values handled per wave

// One wave handles one (b, c-group, 512-t chunk). For each 32-t block:
//   B1 = x at channels {2c,2c+1,96+2c,96+2c+1}, t0..t0+15   (K x N = 4 x 16)
//   B2 = same channels, t0+16..t0+31
//   A1 = W in rows 0-3, A2 = W in rows 8-11 (rest zero)
//   acc = A2*B2 + A1*B1  ==> acc[o] lane L = out channel o at t0+L (full wave)
__global__ __launch_bounds__(256) void mix_kernel(const float* __restrict__ x,
                                                  const float* __restrict__ mask,
                                                  const float* __restrict__ w,
                                                  float* __restrict__ z) {
    const int lane = threadIdx.x & 31;
    const int wid  = blockIdx.x * (blockDim.x >> 5) + (threadIdx.x >> 5);

    const int chunksPerRow  = T_ / CHUNK;          // 8
    const int wavesPerBatch = G_ * chunksPerRow;   // 384
    const int b     = wid / wavesPerBatch;
    const int rem   = wid % wavesPerBatch;
    const int c     = rem / chunksPerRow;
    const int tbase = (rem % chunksPerRow) * CHUNK;

    const int half = lane >> 4;    // 0: lanes 0-15, 1: lanes 16-31
    const int n    = lane & 15;

    // ---- Build A matrices (W is 4x4 row-major: w[o*4+i]) ----
    // A layout (16x4 f32): lane holds row M = lane&15; VGPR0 = K(2*half), VGPR1 = K(2*half+1)
    const int m  = n;
    const int kk = 2 * half;
    v2f a1 = {0.0f, 0.0f};
    v2f a2 = {0.0f, 0.0f};
    if (m < 4)            { a1.x = w[m * 4 + kk];       a1.y = w[m * 4 + kk + 1]; }
    if (m >= 8 && m < 12) { a2.x = w[(m - 8) * 4 + kk]; a2.y = w[(m - 8) * 4 + kk + 1]; }

    // ---- Channel pointers ----
    const int ch0 = 2 * c;            // i=0 / o=0
    const int ch2 = 96 + 2 * c;       // i=2 / o=2
    const int chA = ch0 + half * 96;  // B VGPR0 channel for this lane-half (K=0 or K=2)

    const float* xb = x + (size_t)b * C_ * T_;
    const float* pA = xb + (size_t)chA * T_;        // even channel row (K=0/2)
    const float* pB = pA + T_;                      // odd  channel row (K=1/3)
    const float* pm = mask + (size_t)b * T_;

    float* zb = z + (size_t)b * C_ * T_;
    float* q0 = zb + (size_t)ch0 * T_;
    float* q1 = q0 + T_;
    float* q2 = zb + (size_t)ch2 * T_;
    float* q3 = q2 + T_;

    #pragma unroll 2
    for (int t0 = tbase; t0 < tbase + CHUNK; t0 += 32) {
        // B layout (4x16 f32): lane = column N; VGPR0 = K(2*half), VGPR1 = K(2*half+1)
        v2f b1, b2;
        b1.x = pA[t0 + n];
        b1.y = pB[t0 + n];
        b2.x = pA[t0 + 16 + n];
        b2.y = pB[t0 + 16 + n];
        const float mk = pm[t0 + lane];

        v8f acc = {};
        acc = __builtin_amdgcn_wmma_f32_16x16x4_f32(false, a2, false, b2,
                                                    (short)0, acc, false, false);
        acc = __builtin_amdgcn_wmma_f32_16x16x4_f32(false, a1, false, b1,
                                                    (short)0, acc, false, false);

        // acc[o] @ lane L == output channel o at t = t0 + L  -> fully coalesced stores
        const int t = t0 + lane;
        q0[t] = acc[0] * mk;
        q1[t] = acc[1] * mk;
        q2[t] = acc[2] * mk;
        q3[t] = acc[3] * mk;
    }
}

// logdet[b] = logabsdet(W) * (C/ns) * sum_t(mask[b,t]); one block per batch.
__global__ __launch_bounds__(256) void logdet_kernel(const float* __restrict__ mask,
                                                     const float* __restrict__ w,
                                                     float* __restrict__ out) {
    __shared__ float red[256];
    const int b = blockIdx.x;
    float s = 0.0f;
    for (int t = threadIdx.x; t < T_; t += blockDim.x)
        s += mask[(size_t)b * T_ + t];
    red[threadIdx.x] = s;
    __syncthreads();
    for (int off = 128; off > 0; off >>= 1) {
        if (threadIdx.x < off) red[threadIdx.x] += red[threadIdx.x + off];
        __syncthreads();
    }
    if (threadIdx.x == 0) {
        // slogdet of 4x4 via LU with partial pivoting (sign irrelevant: logabsdet)
        float a[16];
        #pragma unroll
        for (int i = 0; i < 16; ++i) a[i] = w[i];
        float logabs = 0.0f;
        for (int k = 0; k < 4; ++k) {
            int p = k;
            float best = fabsf(a[k * 4 + k]);
            for (int r = k + 1; r < 4; ++r) {
                float v = fabsf(a[r * 4 + k]);
                if (v > best) { best = v; p = r; }
            }
            if (p != k) {
                for (int j = 0; j < 4; ++j) {
                    float tmp = a[k * 4 + j];
                    a[k * 4 + j] = a[p * 4 + j];
                    a[p * 4 + j] = tmp;
                }
            }
            const float piv = a[k * 4 + k];
            logabs += logf(fabsf(piv));
            for (int r = k + 1; r < 4; ++r) {
                const float f = a[r * 4 + k] / piv;
                for (int j = k + 1; j < 4; ++j) a[r * 4 + j] -= f * a[k * 4 + j];
            }
        }
        out[b] = logabs * 48.0f * red[0];
    }
}

extern "C" void kernel_launch(void* const* d_in, const int* in_sizes, int n_in,
                              void* d_out, int out_size, void* d_ws, size_t ws_size,
                              hipStream_t stream) {
    const float* x    = (const float*)d_in[0];   // (16,192,4096) f32
    const float* mask = (const float*)d_in[1];   // (16,1,4096)  f32
    const float* w    = (const float*)d_in[2];   // (4,4)        f32
    float* z  = (float*)d_out;                           // (16,192,4096)
    float* ld = z + (size_t)B_ * C_ * T_;                // (16,)

    // waves = 16 * 48 * (4096/512) = 6144; 8 waves / 256-thread block -> 768 blocks (exact)
    const int totalWaves = B_ * G_ * (T_ / CHUNK);
    mix_kernel<<<totalWaves / 8, 256, 0, stream>>>(x, mask, w, z);
    logdet_kernel<<<B_, 256, 0, stream>>>(mask, w, ld);
}